// EfficientMLHA_64510408786258
// MI455X (gfx1250) — compile-verified
//
#include <hip/hip_runtime.h>
#include <hip/hip_bf16.h>
#include <math.h>

// ---------------- CDNA5 feature detection (also keeps host pass clean) ------
#ifndef __has_builtin
#define __has_builtin(x) 0
#endif

#if __has_builtin(__builtin_amdgcn_wmma_f32_16x16x4_f32)
#define HAS_WMMA_F32 1
#else
#define HAS_WMMA_F32 0
#endif

#if __has_builtin(__builtin_amdgcn_tensor_load_to_lds) && __has_builtin(__builtin_amdgcn_s_wait_tensorcnt)
#define HAS_TDM 1
#else
#define HAS_TDM 0
#endif

typedef float v2f __attribute__((ext_vector_type(2)));
typedef float v8f __attribute__((ext_vector_type(8)));
typedef unsigned int tdm_u4 __attribute__((ext_vector_type(4)));
typedef int tdm_i8 __attribute__((ext_vector_type(8)));
typedef int tdm_i4 __attribute__((ext_vector_type(4)));

#define L_SEQ 2048
#define N_B   16
#define C_IN  256
#define CH3   768
#define KQ    1280          // 256*5
#define LNC   16777216      // 2*L*N*C

__device__ inline v8f wmma_f32_16x16x4(v2f a, v2f b, v8f c) {
#if HAS_WMMA_F32
  // 8 args: (neg_a, A, neg_b, B, c_mod, C, reuse_a, reuse_b)
  return __builtin_amdgcn_wmma_f32_16x16x4_f32(false, a, false, b, (short)0, c, false, false);
#else
  c[0] += a[0] * b[0] + a[1] * b[1];   // placeholder so non-gfx1250 passes compile
  return c;
#endif
}

// ---------------- block reduction (wave32) ----------------------------------
__device__ inline float blockReduceSum(float v, float* sbuf) {
  #pragma unroll
  for (int off = 16; off > 0; off >>= 1) v += __shfl_xor(v, off, 32);
  __syncthreads();
  int wid = threadIdx.x >> 5;
  if ((threadIdx.x & 31) == 0) sbuf[wid] = v;
  __syncthreads();
  int nw = blockDim.x >> 5;
  float r = (threadIdx.x < nw) ? sbuf[threadIdx.x] : 0.f;
  #pragma unroll
  for (int off = 16; off > 0; off >>= 1) r += __shfl_xor(r, off, 32);
  if (threadIdx.x == 0) sbuf[0] = r;
  __syncthreads();
  return sbuf[0];
}

// scale for logmap0(proj(x)) given ||x||^2
__device__ inline float logproj_scale(float ss) {
  float nx = fmaxf(sqrtf(ss), 1e-15f);
  float np = fminf(fminf(nx, 1.f - 1e-5f), 1.f - 1e-7f);
  return atanhf(np) / nx;
}
// scale for logmap0(proj(proj(expmap0(u)))) given ||u||^2  (c=1)
__device__ inline float roundtrip_scale(float ss) {
  float nu = fmaxf(sqrtf(ss), 1e-15f);
  float np = fminf(fminf(tanhf(nu), 1.f - 1e-5f), 1.f - 1e-7f);
  return atanhf(np) / nu;
}

// ---------------- k1: h = logmap0(proj(x)), (L,N,C) -> (N,C,L) --------------
__global__ __launch_bounds__(256) void k1_logmap(const float* __restrict__ x,
                                                 float* __restrict__ h) {
  __shared__ float red[8];
  int b = blockIdx.x, n = b & 15, l = b >> 4;
  int c = threadIdx.x;
  float v = x[((size_t)l * N_B + n) * C_IN + c];
  float ss = blockReduceSum(v * v, red);
  float sc = logproj_scale(ss);
  h[((size_t)n * C_IN + c) * L_SEQ + l] = v * sc;
}

// ---------------- k2: conv_qkv as implicit-im2col GEMM (WMMA f32) -----------
// K ordering is tap-major: k = tap*256 + c, so the (c,tap) split needs no
// integer division and the bound check depends only on (pos+tap), uniform
// across the two K-lanes. Interior waves (all but first/last 64-pos group per
// sequence) take the CHECK=false path: no compares, no EXEC divergence.
template<bool CHECK>
__device__ __forceinline__ void qkv_core(const float* __restrict__ hN,   // h + n*256*L
                                         const float* __restrict__ smemW,
                                         float* __restrict__ yOut,       // y1 + (n*768+ot*16)*L
                                         int p16, int half, int l0) {
  v8f A0 = {}, A1 = {}, A2 = {}, A3 = {};
  int pos0 = l0 + p16;
  #pragma unroll 1
  for (int tap = 0; tap < 5; ++tap) {
    const float* hT = hN + (tap - 2);           // hT[c*L + pos] = h[n,c,pos+tap-2]
    int lpBase = pos0 + tap - 2;
    #pragma unroll 2
    for (int cs = 0; cs < 64; ++cs) {
      int c0 = cs * 4 + 2 * half;               // this lane-group's first channel
      int q0 = p16 * KQ + c0 * 5 + tap;         // LDS tile is in memory K-order
      int q1 = q0 + 5;
      v2f a;
      a.x = smemW[q0 + (q0 >> 8)];
      a.y = smemW[q1 + (q1 >> 8)];
      const float* hc = hT + (size_t)c0 * L_SEQ;
      v2f b0, b1, b2, b3;
      if (CHECK) {
        bool k0 = (unsigned)(lpBase)      < (unsigned)L_SEQ;
        bool k1 = (unsigned)(lpBase + 16) < (unsigned)L_SEQ;
        bool k2 = (unsigned)(lpBase + 32) < (unsigned)L_SEQ;
        bool k3 = (unsigned)(lpBase + 48) < (unsigned)L_SEQ;
        b0.x = k0 ? hc[pos0]              : 0.f;
        b0.y = k0 ? hc[L_SEQ + pos0]      : 0.f;
        b1.x = k1 ? hc[pos0 + 16]         : 0.f;
        b1.y = k1 ? hc[L_SEQ + pos0 + 16] : 0.f;
        b2.x = k2 ? hc[pos0 + 32]         : 0.f;
        b2.y = k2 ? hc[L_SEQ + pos0 + 32] : 0.f;
        b3.x = k3 ? hc[pos0 + 48]         : 0.f;
        b3.y = k3 ? hc[L_SEQ + pos0 + 48] : 0.f;
      } else {
        b0.x = hc[pos0];              b0.y = hc[L_SEQ + pos0];
        b1.x = hc[pos0 + 16];         b1.y = hc[L_SEQ + pos0 + 16];
        b2.x = hc[pos0 + 32];         b2.y = hc[L_SEQ + pos0 + 32];
        b3.x = hc[pos0 + 48];         b3.y = hc[L_SEQ + pos0 + 48];
      }
      A0 = wmma_f32_16x16x4(a, b0, A0);
      A1 = wmma_f32_16x16x4(a, b1, A1);
      A2 = wmma_f32_16x16x4(a, b2, A2);
      A3 = wmma_f32_16x16x4(a, b3, A3);
    }
  }
  #pragma unroll
  for (int r = 0; r < 8; ++r) {
    int m = r + 8 * half;
    float* yr = yOut + (size_t)m * L_SEQ + pos0;
    yr[0]  = A0[r];
    yr[16] = A1[r];
    yr[32] = A2[r];
    yr[48] = A3[r];
  }
}

// grid = 48 outch-tiles * 64 pos-groups; block = 256 (8 waves); wave = 16x64 tile
__global__ __launch_bounds__(256) void k2_conv_qkv(const float* __restrict__ h,
                                                   const float* __restrict__ wq,
                                                   float* __restrict__ y1) {
  __shared__ float smemW[16 * KQ + 80];   // 16 rows x 1280, +1dw pad per 256dw
  int bid = blockIdx.x;
  int ot = bid >> 6;                      // out-channel tile [0,48)
  int pg = bid & 63;
  int n = pg >> 2;
  int lbase = (pg & 3) * 512;

#if HAS_TDM
  if ((threadIdx.x >> 5) == 0) {
    unsigned long long ga = (unsigned long long)(size_t)(const void*)(wq + (size_t)ot * 16 * KQ);
    unsigned ldsOff = (unsigned)(size_t)(const void*)&smemW[0];   // LDS aperture: addr[31:0] = offset
    // D# group0: count=1 | lds_addr | global_addr[56:0] | type=2
    tdm_u4 g0 = { 1u, ldsOff, (unsigned)ga,
                  (unsigned)((ga >> 32) & 0x1FFFFFFu) | 0x80000000u };
    // D# group1: data_size=4B, pad_enable, pad_interval=256dw, pad_amount=1dw,
    //            tensor_dim0=1280, tensor_dim1=16, tile_dim0=1280, tile_dim1=16, stride0=1280
    tdm_i8 g1 = { (int)((2u << 16) | (1u << 20) | (7u << 22)),
                  (int)((KQ & 0xFFFFu) << 16),
                  (int)(((KQ >> 16) & 0xFFFFu) | (16u << 16)),
                  (int)(((16u >> 16) & 0xFFFFu) | ((KQ & 0xFFFFu) << 16)),
                  (int)16,
                  (int)KQ,
                  0, 0 };
    tdm_i4 g2 = { 0, 0, 0, 0 };
    tdm_i4 g3 = { 0, 0, 0, 0 };
    tdm_i8 g4 = { 0, 0, 0, 0, 0, 0, 0, 0 };
    __builtin_amdgcn_tensor_load_to_lds(g0, g1, g2, g3, g4, 0);
    __builtin_amdgcn_s_wait_tensorcnt(0);
  }
#else
  for (int i = threadIdx.x; i < 16 * KQ; i += 256)
    smemW[i + (i >> 8)] = wq[(size_t)ot * 16 * KQ + i];
#endif
  __syncthreads();

  int lane = threadIdx.x & 31;
  int wv = threadIdx.x >> 5;
  int half = lane >> 4, p16 = lane & 15;
  int l0 = lbase + wv * 64;

  const float* hN = h + (size_t)n * C_IN * L_SEQ;
  float* yOut = y1 + ((size_t)n * CH3 + ot * 16) * L_SEQ;
  bool interior = (l0 >= 2) && (l0 + 66 <= L_SEQ);   // wave-uniform
  if (interior) qkv_core<false>(hN, smemW, yOut, p16, half, l0);
  else          qkv_core<true >(hN, smemW, yOut, p16, half, l0);
}

// ---------------- k3: BN stats per channel ----------------------------------
__global__ __launch_bounds__(256) void k3_bnstats(const float* __restrict__ y1,
                                                  const float* __restrict__ gamma,
                                                  const float* __restrict__ beta,
                                                  float* __restrict__ stats) {
  __shared__ float red[8];
  int ch = blockIdx.x;
  float s1 = 0.f, s2 = 0.f;
  for (int i = threadIdx.x; i < N_B * L_SEQ; i += 256) {
    int n = i >> 11, l = i & 2047;
    float v = y1[((size_t)n * CH3 + ch) * L_SEQ + l];
    s1 += v; s2 += v * v;
  }
  s1 = blockReduceSum(s1, red);
  s2 = blockReduceSum(s2, red);
  if (threadIdx.x == 0) {
    float mean = s1 / 32768.f;
    float var  = s2 / 32768.f - mean * mean;
    float a = gamma[ch] * rsqrtf(var + 1e-5f);
    stats[ch] = a;
    stats[CH3 + ch] = beta[ch] - mean * a;
  }
}

// ---------------- k4: BN+relu+hyperbolic round trip (in place) --------------
__global__ __launch_bounds__(256) void k4_postconv(float* __restrict__ t,
                                                   const float* __restrict__ stats) {
  __shared__ float red[8];
  int b = blockIdx.x, n = b & 15, l = b >> 4;
  float vv[3]; float ss = 0.f;
  #pragma unroll
  for (int r = 0; r < 3; ++r) {
    int ch = threadIdx.x + 256 * r;
    float y = t[((size_t)n * CH3 + ch) * L_SEQ + l];
    float v = fmaxf(stats[ch] * y + stats[CH3 + ch], 0.f);
    vv[r] = v; ss += v * v;
  }
  float sc = roundtrip_scale(blockReduceSum(ss, red));
  #pragma unroll
  for (int r = 0; r < 3; ++r) {
    int ch = threadIdx.x + 256 * r;
    t[((size_t)n * CH3 + ch) * L_SEQ + l] = vv[r] * sc;
  }
}

// ---------------- k5: depthwise conv k=5 ------------------------------------
__global__ __launch_bounds__(256) void k5_dw(const float* __restrict__ t,
                                             const float* __restrict__ wdw,
                                             float* __restrict__ o) {
  int idx = blockIdx.x * 256 + threadIdx.x;         // (n,ch,l) linear
  int l = idx & 2047; int nc = idx >> 11;
  int ch = nc % CH3;
  const float* row = t + (size_t)nc * L_SEQ;
  const float* w = wdw + ch * 5;
  float s = 0.f;
  #pragma unroll
  for (int kk = 0; kk < 5; ++kk) {
    int lp = l + kk - 2;
    s += ((unsigned)lp < (unsigned)L_SEQ) ? row[lp] * w[kk] : 0.f;
  }
  o[idx] = s;
}

// ---------------- k6: grouped pointwise (24x 32x32) via WMMA f32 ------------
// grid = N * 24 * 32 ; block = 32 (1 wave) computing 32 outch x 64 pos
__global__ __launch_bounds__(32) void k6_pointwise(const float* __restrict__ dw,
                                                   const float* __restrict__ wpw,
                                                   float* __restrict__ s) {
  int bid = blockIdx.x;
  int n = bid / (24 * 32);
  int rem = bid % (24 * 32);
  int g = rem >> 5;
  int l0 = (rem & 31) * 64;
  int lane = threadIdx.x, half = lane >> 4, p16 = lane & 15;
  int pos0 = l0 + p16;

  v8f A0 = {}, A1 = {}, A2 = {}, A3 = {};
  v8f B0 = {}, B1 = {}, B2 = {}, B3 = {};
  #pragma unroll
  for (int ks = 0; ks < 8; ++ks) {
    int kb = ks * 4 + 2 * half;
    v2f a0, a1;
    a0.x = wpw[((size_t)g * 32 + p16) * 32 + kb];
    a0.y = wpw[((size_t)g * 32 + p16) * 32 + kb + 1];
    a1.x = wpw[((size_t)g * 32 + 16 + p16) * 32 + kb];
    a1.y = wpw[((size_t)g * 32 + 16 + p16) * 32 + kb + 1];
    const float* db = dw + ((size_t)n * CH3 + g * 32 + kb) * L_SEQ;
    v2f b0, b1, b2, b3;
    b0.x = db[pos0];      b0.y = db[L_SEQ + pos0];
    b1.x = db[pos0 + 16]; b1.y = db[L_SEQ + pos0 + 16];
    b2.x = db[pos0 + 32]; b2.y = db[L_SEQ + pos0 + 32];
    b3.x = db[pos0 + 48]; b3.y = db[L_SEQ + pos0 + 48];
    A0 = wmma_f32_16x16x4(a0, b0, A0);
    A1 = wmma_f32_16x16x4(a0, b1, A1);
    A2 = wmma_f32_16x16x4(a0, b2, A2);
    A3 = wmma_f32_16x16x4(a0, b3, A3);
    B0 = wmma_f32_16x16x4(a1, b0, B0);
    B1 = wmma_f32_16x16x4(a1, b1, B1);
    B2 = wmma_f32_16x16x4(a1, b2, B2);
    B3 = wmma_f32_16x16x4(a1, b3, B3);
  }
  #pragma unroll
  for (int r = 0; r < 8; ++r) {
    int m = r + 8 * half;
    float* s0 = s + ((size_t)n * CH3 + g * 32 + m) * L_SEQ + pos0;
    float* s1 = s + ((size_t)n * CH3 + g * 32 + 16 + m) * L_SEQ + pos0;
    s0[0] = A0[r]; s0[16] = A1[r]; s0[32] = A2[r]; s0[48] = A3[r];
    s1[0] = B0[r]; s1[16] = B1[r]; s1[32] = B2[r]; s1[48] = B3[r];
  }
}

// ---------------- k7: concat round-trip over 1536 channels (in place) -------
__global__ __launch_bounds__(256) void k7_cat(float* __restrict__ t,
                                              float* __restrict__ s) {
  __shared__ float red[8];
  int b = blockIdx.x, n = b & 15, l = b >> 4;
  float vt[3], vs[3]; float ss = 0.f;
  #pragma unroll
  for (int r = 0; r < 3; ++r) {
    int ch = threadIdx.x + 256 * r;
    size_t off = ((size_t)n * CH3 + ch) * L_SEQ + l;
    vt[r] = t[off]; vs[r] = s[off];
    ss += vt[r] * vt[r] + vs[r] * vs[r];
  }
  float sc = roundtrip_scale(blockReduceSum(ss, red));
  #pragma unroll
  for (int r = 0; r < 3; ++r) {
    int ch = threadIdx.x + 256 * r;
    size_t off = ((size_t)n * CH3 + ch) * L_SEQ + l;
    t[off] = vt[r] * sc; s[off] = vs[r] * sc;
  }
}

// ---------------- k8: linear attention (einsum collapses to reductions) -----
// out[d,l] = relu(q[d,l]) * (sum_l v[d,l]*ksum[l]) / (sum_l v[d,l] + eps)
__global__ __launch_bounds__(256) void k8_attn(const float* __restrict__ t2a,
                                               const float* __restrict__ t2b,
                                               float* __restrict__ ob) {
  __shared__ float ksum[L_SEQ];
  __shared__ float red[256];
  __shared__ float fac[32];
  int b = blockIdx.x, n = b >> 4, g = b & 15;
  const float* base = (g < 8) ? (t2a + ((size_t)n * CH3 + g * 96) * L_SEQ)
                              : (t2b + ((size_t)n * CH3 + (g - 8) * 96) * L_SEQ);
  const float* Q = base;
  const float* K = base + 32 * L_SEQ;
  const float* V = base + 64 * L_SEQ;
  int tid = threadIdx.x;
  for (int l = tid; l < L_SEQ; l += 256) {
    float s = 0.f;
    for (int e = 0; e < 32; ++e) s += fmaxf(K[e * L_SEQ + l], 0.f);
    ksum[l] = s;
  }
  __syncthreads();
  int d = tid >> 3, ck = tid & 7;
  float vk = 0.f, vsum = 0.f;
  for (int l = ck * 256; l < ck * 256 + 256; ++l) {
    float vv = V[d * L_SEQ + l];
    vk += vv * ksum[l]; vsum += vv;
  }
  red[tid] = vk; __syncthreads();
  if (ck == 0) { float s = 0.f; for (int j = 0; j < 8; ++j) s += red[tid + j]; fac[d] = s; }
  __syncthreads();
  red[tid] = vsum; __syncthreads();
  if (ck == 0) { float s = 0.f; for (int j = 0; j < 8; ++j) s += red[tid + j]; fac[d] = fac[d] / (s + 1e-15f); }
  __syncthreads();
  for (int i = tid; i < 32 * L_SEQ; i += 256) {
    int dd = i >> 11, l = i & 2047;
    ob[((size_t)n * 512 + g * 32 + dd) * L_SEQ + l] = fmaxf(Q[dd * L_SEQ + l], 0.f) * fac[dd];
  }
}

// ---------------- k9: round-trip over 512 channels (in place) ---------------
__global__ __launch_bounds__(256) void k9_outchain(float* __restrict__ a) {
  __shared__ float red[8];
  int b = blockIdx.x, n = b & 15, l = b >> 4;
  float v0 = a[((size_t)n * 512 + threadIdx.x) * L_SEQ + l];
  float v1 = a[((size_t)n * 512 + threadIdx.x + 256) * L_SEQ + l];
  float sc = roundtrip_scale(blockReduceSum(v0 * v0 + v1 * v1, red));
  a[((size_t)n * 512 + threadIdx.x) * L_SEQ + l] = v0 * sc;
  a[((size_t)n * 512 + threadIdx.x + 256) * L_SEQ + l] = v1 * sc;
}

// ---------------- k10: conv_res 1x1 (256x512 GEMM) via WMMA f32 -------------
__global__ __launch_bounds__(32) void k10_convres(const float* __restrict__ ft,
                                                  const float* __restrict__ wr,
                                                  const float* __restrict__ br,
                                                  float* __restrict__ ob) {
  int bid = blockIdx.x;
  int ot = bid >> 9;                 // 16 outch tiles
  int pgr = bid & 511;
  int n = pgr >> 5;
  int l0 = (pgr & 31) * 64;
  int lane = threadIdx.x, half = lane >> 4, p16 = lane & 15;
  int pos0 = l0 + p16;

  v8f A0 = {}, A1 = {}, A2 = {}, A3 = {};
  #pragma unroll 4
  for (int ks = 0; ks < 128; ++ks) {
    int kb = ks * 4 + 2 * half;
    v2f a;
    a.x = wr[((size_t)(ot * 16 + p16)) * 512 + kb];
    a.y = wr[((size_t)(ot * 16 + p16)) * 512 + kb + 1];
    const float* fb = ft + ((size_t)n * 512 + kb) * L_SEQ;
    v2f b0, b1, b2, b3;
    b0.x = fb[pos0];      b0.y = fb[L_SEQ + pos0];
    b1.x = fb[pos0 + 16]; b1.y = fb[L_SEQ + pos0 + 16];
    b2.x = fb[pos0 + 32]; b2.y = fb[L_SEQ + pos0 + 32];
    b3.x = fb[pos0 + 48]; b3.y = fb[L_SEQ + pos0 + 48];
    A0 = wmma_f32_16x16x4(a, b0, A0);
    A1 = wmma_f32_16x16x4(a, b1, A1);
    A2 = wmma_f32_16x16x4(a, b2, A2);
    A3 = wmma_f32_16x16x4(a, b3, A3);
  }
  #pragma unroll
  for (int r = 0; r < 8; ++r) {
    int m = r + 8 * half;
    float bias = br[ot * 16 + m];
    float* o0 = ob + ((size_t)n * C_IN + ot * 16 + m) * L_SEQ + pos0;
    o0[0]  = A0[r] + bias;
    o0[16] = A1[r] + bias;
    o0[32] = A2[r] + bias;
    o0[48] = A3[r] + bias;
  }
}

// ---------------- k11: fx = proj(expmap0(ft)); residual = mobius_add --------
__global__ __launch_bounds__(256) void k11_final(const float* __restrict__ ftb,
                                                 const float* __restrict__ x,
                                                 float* __restrict__ out) {
  __shared__ float red[8];
  int b = blockIdx.x, n = b & 15, l = b >> 4;
  int c = threadIdx.x;
  float ftv = ftb[((size_t)n * C_IN + c) * L_SEQ + l];
  float xv = x[((size_t)l * N_B + n) * C_IN + c];
  float sff = blockReduceSum(ftv * ftv, red);
  float sfx = blockReduceSum(ftv * xv, red);
  float sxx = blockReduceSum(xv * xv, red);
  float nf = fmaxf(sqrtf(sff), 1e-15f);
  float sc = fminf(tanhf(nf), 1.f - 1e-5f) / nf;    // expmap0 then proj clip
  float fx = sc * ftv;
  float x2 = sc * sc * sff;                          // |fx|^2
  float xy = sc * sfx;                               // fx . x
  float y2 = sxx;                                    // |x|^2
  float num = (1.f + 2.f * xy + y2) * fx + (1.f - x2) * xv;
  float den = fmaxf(1.f + 2.f * xy + x2 * y2, 1e-15f);
  size_t o = ((size_t)l * N_B + n) * C_IN + c;
  out[o] = fx;
  out[LNC + o] = num / den;
}

// ---------------- launch -----------------------------------------------------
extern "C" void kernel_launch(void* const* d_in, const int* in_sizes, int n_in,
                              void* d_out, int out_size, void* d_ws, size_t ws_size,
                              hipStream_t stream) {
  const float* x     = (const float*)d_in[0];
  const float* w_qkv = (const float*)d_in[1];
  const float* gam   = (const float*)d_in[2];
  const float* bet   = (const float*)d_in[3];
  const float* w_dw  = (const float*)d_in[4];
  const float* w_pw  = (const float*)d_in[5];
  const float* w_res = (const float*)d_in[6];
  const float* b_res = (const float*)d_in[7];
  float* out = (float*)d_out;

  float* ws = (float*)d_ws;
  float* bufH   = ws;                      // (N,256,L)   8,388,608
  float* bufT   = bufH + 8388608;          // (N,768,L)  25,165,824  conv out -> t -> t2a
  float* bufDW  = bufT + 25165824;         // (N,768,L)  depthwise out
  float* bufS   = bufDW + 25165824;        // (N,768,L)  pointwise out -> t2b
  float* bufATT = bufS + 25165824;         // (N,512,L)  attn out -> ft
  float* stats  = bufATT + 16777216;       // 1536
  float* bufR   = bufH;                    // (N,256,L)  conv_res out (reuse h)

  k1_logmap   <<<32768, 256, 0, stream>>>(x, bufH);
  k2_conv_qkv <<<48 * 64, 256, 0, stream>>>(bufH, w_qkv, bufT);
  k3_bnstats  <<<CH3, 256, 0, stream>>>(bufT, gam, bet, stats);
  k4_postconv <<<32768, 256, 0, stream>>>(bufT, stats);
  k5_dw       <<<98304, 256, 0, stream>>>(bufT, w_dw, bufDW);
  k6_pointwise<<<16 * 24 * 32, 32, 0, stream>>>(bufDW, w_pw, bufS);
  k7_cat      <<<32768, 256, 0, stream>>>(bufT, bufS);
  k8_attn     <<<256, 256, 0, stream>>>(bufT, bufS, bufATT);
  k9_outchain <<<32768, 256, 0, stream>>>(bufATT);
  k10_convres <<<16 * 512, 32, 0, stream>>>(bufATT, w_res, b_res, bufR);
  k11_final   <<<32768, 256, 0, stream>>>(bufR, x, out);

  (void)in_sizes; (void)n_in; (void)out_size; (void)ws_size;
}